// GSNN_89326729822575
// MI455X (gfx1250) — compile-verified
//
#include <hip/hip_runtime.h>
#include <cstddef>

#define NN   10000
#define EE   160000
#define FF   8000
#define HH   64000
#define BB   128
#define LLAY 4
#define E1   140000
#define E0   20000
#define NW1  1120000
#define NW2  512000
#define NW3  1120000
#define LN_EPS 1e-5f

typedef float v2f __attribute__((ext_vector_type(2)));
typedef float v8f __attribute__((ext_vector_type(8)));

// Branch-free ELU: native v_exp_f32 on clamped arg + v_cndmask select.
__device__ __forceinline__ float elu_f(float x) {
    float ex = __expf(fminf(x, 0.f)) - 1.f;
    return x > 0.f ? x : ex;
}

// ---------------- utility kernels ----------------
__global__ void k_zero_i(int* p, int n) {
    int i = blockIdx.x * 256 + threadIdx.x;
    if (i < n) p[i] = 0;
}
__global__ void k_zero_f(float* p, int n) {
    int i = blockIdx.x * 256 + threadIdx.x;
    if (i < n) p[i] = 0.f;
}
__global__ void k_copy_i(const int* __restrict__ a, int* __restrict__ b, int n) {
    int i = blockIdx.x * 256 + threadIdx.x;
    if (i < n) b[i] = a[i];
}

// ---------------- CSR build (deterministic: atomic fill + per-node sort) ----------------
__global__ void k_count1(const int* __restrict__ dst, int* cnt) {
    int e = blockIdx.x * 256 + threadIdx.x;
    if (e < E1) atomicAdd(&cnt[dst[e] - 1000], 1);
}
__global__ void k_count2(const int* __restrict__ dst, int* cnt) {
    int e = blockIdx.x * 256 + threadIdx.x;
    if (e < EE) { int d = dst[e]; if (d >= 9000) atomicAdd(&cnt[d - 9000], 1); }
}
__global__ void k_fill1(const int* __restrict__ dst, int* cur, int* lst) {
    int e = blockIdx.x * 256 + threadIdx.x;
    if (e < E1) { int p = atomicAdd(&cur[dst[e] - 1000], 1); lst[p] = e; }
}
__global__ void k_fill2(const int* __restrict__ dst, int* cur, int* lst) {
    int e = blockIdx.x * 256 + threadIdx.x;
    if (e < EE) { int d = dst[e]; if (d >= 9000) { int p = atomicAdd(&cur[d - 9000], 1); lst[p] = e; } }
}
// single-block exclusive scan, n <= 8192
__global__ void k_scan(const int* __restrict__ cnt, int* __restrict__ off, int n) {
    __shared__ int part[1024];
    int t = threadIdx.x;
    int ipt = (n + 1023) / 1024;
    int base = t * ipt;
    int s = 0;
    for (int k = 0; k < ipt; ++k) { int i = base + k; if (i < n) s += cnt[i]; }
    part[t] = s;
    __syncthreads();
    for (int ofs = 1; ofs < 1024; ofs <<= 1) {
        int v = part[t];
        int add = (t >= ofs) ? part[t - ofs] : 0;
        __syncthreads();
        part[t] = v + add;
        __syncthreads();
    }
    int run = (t == 0) ? 0 : part[t - 1];
    for (int k = 0; k < ipt; ++k) {
        int i = base + k;
        if (i < n) { off[i] = run; run += cnt[i]; }
    }
    if (t == 1023) off[n] = part[1023];
}
__global__ void k_sort(const int* __restrict__ off, int* lst, int n) {
    int v = blockIdx.x * 256 + threadIdx.x;
    if (v >= n) return;
    int s = off[v], e = off[v + 1];
    for (int i = s + 1; i < e; ++i) {
        int key = lst[i]; int j = i - 1;
        while (j >= s && lst[j] > key) { lst[j + 1] = lst[j]; --j; }
        lst[j + 1] = key;
    }
}

// ---------------- data movement ----------------
// xt[n*B + b] = x[b*N + n]  (LDS tiled transpose)
__global__ void k_transpose(const float* __restrict__ x, float* __restrict__ xt) {
    __shared__ float tile[32][33];
    int n0 = blockIdx.x * 32, b0 = blockIdx.y * 32;
    int tx = threadIdx.x, ty = threadIdx.y;
    int n = n0 + tx;
    if (n < NN) tile[ty][tx] = x[(size_t)(b0 + ty) * NN + n];
    __syncthreads();
    int nw = n0 + ty;
    if (nw < NN) xt[(size_t)nw * BB + b0 + tx] = tile[tx][ty];
}
// h[e][b] = x0 = xt[src[e]][b]
__global__ void k_hinit(const float* __restrict__ xt, const int* __restrict__ src, float* __restrict__ h) {
    int idx = blockIdx.x * 256 + threadIdx.x;
    if (idx >= EE * BB) return;
    int e = idx >> 7, b = idx & 127;
    h[idx] = xt[(size_t)src[e] * BB + b];
}

// ---------------- layer kernels ----------------
// a_hid[(v*8+c)][b] = elu( sum_{e in in(v)} w1[e*8+c] * h[e][b] + b1[v*8+c] )
__global__ void k_spmm1(const float* __restrict__ h, const float* __restrict__ w1l,
                        const float* __restrict__ b1l, const int* __restrict__ off,
                        const int* __restrict__ lst, float* __restrict__ ah) {
    int v = blockIdx.x;
    int b = threadIdx.x;
    float acc[8] = {0.f,0.f,0.f,0.f,0.f,0.f,0.f,0.f};
    int s = off[v], e_end = off[v + 1];
    for (int i = s; i < e_end; ++i) {
        int e = lst[i];
        float hv = h[(size_t)e * BB + b];
        const float4* wp = (const float4*)(w1l + (size_t)e * 8);
        float4 wa = wp[0], wb = wp[1];
        acc[0] += wa.x * hv; acc[1] += wa.y * hv; acc[2] += wa.z * hv; acc[3] += wa.w * hv;
        acc[4] += wb.x * hv; acc[5] += wb.y * hv; acc[6] += wb.z * hv; acc[7] += wb.w * hv;
    }
    #pragma unroll
    for (int c = 0; c < 8; ++c)
        ah[(size_t)(v * 8 + c) * BB + b] = elu_f(acc[c] + b1l[v * 8 + c]);
}

// Per-node 8x8 channel mix over 128 batch via V_WMMA_F32_16X16X4_F32, in-place, fused bias+ELU.
// D(16x16) = A(16x4) x B(4x16) + C ; M rows 0..7 = output channel j, N = 16-batch tile, K = input channel.
// 32-bit A layout: lanes 0-15 M=0..15; VGPR0 holds K=0 (low half)/K=2 (high half), VGPR1 K=1/K=3.
// 32-bit B layout: VGPR0 row K=0 (lanes0-15)/K=2 (lanes16-31), VGPR1 K=1/K=3.
__global__ void k_spmm2_wmma(float* __restrict__ ah, const float* __restrict__ w2l,
                             const float* __restrict__ b2l) {
    int lane = threadIdx.x & 31;
    int wv   = threadIdx.x >> 5;
    int f    = blockIdx.x * 8 + wv;         // hidden node, one wave per node
    int j    = lane & 15;
    int half = lane >> 4;
    bool jv  = (j < 8);
    const float* wf = w2l + (size_t)f * 64; // W2[i][j] = wf[i*8+j]
    v2f A0, A1;                             // A[m=j][k=i] = W2[i][j]
    A0.x = jv ? wf[(2 * half + 0) * 8 + j] : 0.f;
    A0.y = jv ? wf[(2 * half + 1) * 8 + j] : 0.f;
    A1.x = jv ? wf[(4 + 2 * half + 0) * 8 + j] : 0.f;
    A1.y = jv ? wf[(4 + 2 * half + 1) * 8 + j] : 0.f;
    float bb[8];
    #pragma unroll
    for (int r = 0; r < 8; ++r) bb[r] = b2l[f * 8 + r];
    float* base = ah + (size_t)f * 8 * BB;
    #pragma unroll 1
    for (int mt = 0; mt < 8; ++mt) {
        int b0 = mt * 16;
        v2f B0, B1;                         // B[k=i][n] = ah[(f*8+i)][b0+n]
        B0.x = base[(2 * half + 0) * BB + b0 + j];
        B0.y = base[(2 * half + 1) * BB + b0 + j];
        B1.x = base[(4 + 2 * half + 0) * BB + b0 + j];
        B1.y = base[(4 + 2 * half + 1) * BB + b0 + j];
        v8f c = {0.f,0.f,0.f,0.f,0.f,0.f,0.f,0.f};
        c = __builtin_amdgcn_wmma_f32_16x16x4_f32(false, A0, false, B0, (short)0, c, false, false);
        c = __builtin_amdgcn_wmma_f32_16x16x4_f32(false, A1, false, B1, (short)0, c, false, false);
        if (lane < 16) {                    // D vgpr r, lanes 0-15: M=r (=out channel), N=lane (=batch)
            #pragma unroll
            for (int r = 0; r < 8; ++r)
                base[r * BB + b0 + j] = elu_f(c[r] + bb[r]);
        }
    }
}

// ae[e][b] = h[e][b] + b3[e] + (e>=E0 ? sum_c w3[(e-E0)*8+c] * ah[(src-1000)*8+c][b] : 0)
__global__ void k_spmm3(const float* __restrict__ h, const float* __restrict__ ah,
                        const float* __restrict__ w3l, const float* __restrict__ b3l,
                        const int* __restrict__ src, float* __restrict__ ae) {
    int e = blockIdx.x;
    int b = threadIdx.x;
    float v = h[(size_t)e * BB + b] + b3l[e];
    if (e >= E0) {
        int hb = (src[e] - 1000) * 8;
        const float4* wp = (const float4*)(w3l + (size_t)(e - E0) * 8);
        float4 wa = wp[0], wb = wp[1];
        const float* ap = ah + (size_t)hb * BB + b;
        v += wa.x * ap[0 * BB]; v += wa.y * ap[1 * BB];
        v += wa.z * ap[2 * BB]; v += wa.w * ap[3 * BB];
        v += wb.x * ap[4 * BB]; v += wb.y * ap[5 * BB];
        v += wb.z * ap[6 * BB]; v += wb.w * ap[7 * BB];
    }
    ae[(size_t)e * BB + b] = v;
}

// layer-norm stats: 128 e-strips x 128 batch lanes, then final reduce
__global__ void k_lnstats1(const float* __restrict__ ae, float* __restrict__ ps, float* __restrict__ pss) {
    int blk = blockIdx.x;      // 0..127
    int b = threadIdx.x;
    size_t e0 = (size_t)blk * (EE / 128);
    float s = 0.f, ss = 0.f;
    for (int k = 0; k < EE / 128; ++k) {
        float v = ae[(e0 + k) * BB + b];
        s += v; ss += v * v;
    }
    ps[blk * BB + b] = s;
    pss[blk * BB + b] = ss;
}
__global__ void k_lnstats2(const float* __restrict__ ps, const float* __restrict__ pss,
                           float* __restrict__ mean, float* __restrict__ rstd) {
    int b = threadIdx.x;
    float s = 0.f, ss = 0.f;
    for (int i = 0; i < 128; ++i) { s += ps[i * BB + b]; ss += pss[i * BB + b]; }
    float mu = s / (float)EE;
    float var = ss / (float)EE - mu * mu;
    mean[b] = mu;
    rstd[b] = rsqrtf(var + LN_EPS);
}

// h = iem ? x0 : (ae - mu) * rstd
__global__ void k_norm(const float* __restrict__ ae, const float* __restrict__ mean,
                       const float* __restrict__ rstd, const int* __restrict__ src,
                       const int* __restrict__ inm, const float* __restrict__ xt,
                       float* __restrict__ h) {
    int idx = blockIdx.x * 256 + threadIdx.x;
    if (idx >= EE * BB) return;
    int e = idx >> 7, b = idx & 127;
    int sn = src[e];
    if (inm[sn]) h[idx] = xt[(size_t)sn * BB + b];
    else         h[idx] = (ae[idx] - mean[b]) * rstd[b];
}

// out[b][9000+n] = outm * sum_{e in out(n)} (scale[e]*h[e][b] + bias[e])
__global__ void k_out(const float* __restrict__ h, const float* __restrict__ scale,
                      const float* __restrict__ bias, const int* __restrict__ off,
                      const int* __restrict__ lst, const int* __restrict__ outm,
                      float* __restrict__ out) {
    int n = blockIdx.x;   // output node 9000+n
    int b = threadIdx.x;
    float acc = 0.f;
    int s = off[n], e_end = off[n + 1];
    for (int i = s; i < e_end; ++i) {
        int e = lst[i];
        acc += scale[e] * h[(size_t)e * BB + b] + bias[e];
    }
    out[(size_t)b * NN + 9000 + n] = acc * (float)outm[9000 + n];
}

// ---------------- host launcher ----------------
extern "C" void kernel_launch(void* const* d_in, const int* in_sizes, int n_in,
                              void* d_out, int out_size, void* d_ws, size_t ws_size,
                              hipStream_t stream) {
    (void)in_sizes; (void)n_in; (void)out_size; (void)ws_size;
    const float* x     = (const float*)d_in[0];
    const float* w1    = (const float*)d_in[1];
    const float* b1    = (const float*)d_in[2];
    const float* w2    = (const float*)d_in[3];
    const float* b2    = (const float*)d_in[4];
    const float* w3    = (const float*)d_in[5];
    const float* b3    = (const float*)d_in[6];
    const float* scl   = (const float*)d_in[7];
    const float* bia   = (const float*)d_in[8];
    const int*   src   = (const int*)d_in[15];
    const int*   dst   = (const int*)d_in[16];
    const int*   inm   = (const int*)d_in[17];
    const int*   outm  = (const int*)d_in[18];
    float* out = (float*)d_out;

    char* w = (char*)d_ws;
    size_t o = 0;
    auto alloc = [&](size_t bytes) -> void* { void* p = w + o; o = (o + bytes + 255) & ~(size_t)255; return p; };
    float* xt   = (float*)alloc(sizeof(float) * (size_t)NN * BB);
    float* h    = (float*)alloc(sizeof(float) * (size_t)EE * BB);
    float* ae   = (float*)alloc(sizeof(float) * (size_t)EE * BB);
    float* ah   = (float*)alloc(sizeof(float) * (size_t)HH * BB);
    float* ps   = (float*)alloc(sizeof(float) * 128 * BB);
    float* pss  = (float*)alloc(sizeof(float) * 128 * BB);
    float* mean = (float*)alloc(sizeof(float) * BB);
    float* rstd = (float*)alloc(sizeof(float) * BB);
    int* cnt1 = (int*)alloc(sizeof(int) * FF);
    int* off1 = (int*)alloc(sizeof(int) * (FF + 1));
    int* cur1 = (int*)alloc(sizeof(int) * FF);
    int* lst1 = (int*)alloc(sizeof(int) * E1);
    int* cnt2 = (int*)alloc(sizeof(int) * 1000);
    int* off2 = (int*)alloc(sizeof(int) * 1001);
    int* cur2 = (int*)alloc(sizeof(int) * 1000);
    int* lst2 = (int*)alloc(sizeof(int) * E0);

    // Build CSRs (every call; deterministic thanks to per-node sort)
    k_zero_i<<<(FF + 255) / 256, 256, 0, stream>>>(cnt1, FF);
    k_zero_i<<<(1000 + 255) / 256, 256, 0, stream>>>(cnt2, 1000);
    k_count1<<<(E1 + 255) / 256, 256, 0, stream>>>(dst, cnt1);
    k_count2<<<(EE + 255) / 256, 256, 0, stream>>>(dst, cnt2);
    k_scan<<<1, 1024, 0, stream>>>(cnt1, off1, FF);
    k_scan<<<1, 1024, 0, stream>>>(cnt2, off2, 1000);
    k_copy_i<<<(FF + 255) / 256, 256, 0, stream>>>(off1, cur1, FF);
    k_copy_i<<<(1000 + 255) / 256, 256, 0, stream>>>(off2, cur2, 1000);
    k_fill1<<<(E1 + 255) / 256, 256, 0, stream>>>(dst, cur1, lst1);
    k_fill2<<<(EE + 255) / 256, 256, 0, stream>>>(dst, cur2, lst2);
    k_sort<<<(FF + 255) / 256, 256, 0, stream>>>(off1, lst1, FF);
    k_sort<<<(1000 + 255) / 256, 256, 0, stream>>>(off2, lst2, 1000);

    // x -> batch-minor, h = x0
    dim3 tb(32, 32);
    dim3 tg((NN + 31) / 32, BB / 32);
    k_transpose<<<tg, tb, 0, stream>>>(x, xt);
    k_hinit<<<(EE * BB + 255) / 256, 256, 0, stream>>>(xt, src, h);

    for (int l = 0; l < LLAY; ++l) {
        k_spmm1<<<FF, BB, 0, stream>>>(h, w1 + (size_t)l * NW1, b1 + (size_t)l * HH, off1, lst1, ah);
        k_spmm2_wmma<<<FF / 8, 256, 0, stream>>>(ah, w2 + (size_t)l * NW2, b2 + (size_t)l * HH);
        k_spmm3<<<EE, BB, 0, stream>>>(h, ah, w3 + (size_t)l * NW3, b3 + (size_t)l * EE, src, ae);
        k_lnstats1<<<128, BB, 0, stream>>>(ae, ps, pss);
        k_lnstats2<<<1, BB, 0, stream>>>(ps, pss, mean, rstd);
        k_norm<<<(EE * BB + 255) / 256, 256, 0, stream>>>(ae, mean, rstd, src, inm, xt, h);
    }

    k_zero_f<<<(BB * NN + 255) / 256, 256, 0, stream>>>(out, BB * NN);
    k_out<<<1000, BB, 0, stream>>>(h, scl, bia, off2, lst2, outm, out);
}